// MappedMaxPool_34282428956676
// MI455X (gfx1250) — compile-verified
//
#include <hip/hip_runtime.h>
#include <hip/hip_bf16.h>
#include <stdint.h>

// Problem constants (from reference)
#define B_    4
#define C_    64
#define H_    256
#define W_    512
#define OH_   128
#define OW_   256
#define SITES (OH_ * OW_)          // 32768
#define HW_   ((size_t)H_ * W_)    // 131072
#define BC_   256

// ---------------------------------------------------------------------------
// Generic batched 64x64 tiled transpose:  in (N, R, C)  ->  out (N, C, R)
// Uses gfx1250 async global<->LDS ops: data flows HBM -> LDS -> HBM without
// passing through VGPRs. LDS tile padded to 65 floats -> conflict-free in
// both phases (phase1 banks = lane, phase2 bank stride 65 == odd).
// ---------------------------------------------------------------------------
__global__ __launch_bounds__(256) void transpose64(const float* __restrict__ in,
                                                   float* __restrict__ out,
                                                   int R, int Ccols) {
  __shared__ float tile[64][65];
  const int t  = threadIdx.x;
  const int lx = t & 63;   // fast dim within tile
  const int ly = t >> 6;   // 0..3
  const size_t mat = (size_t)R * (size_t)Ccols;
  const uint64_t src = (uint64_t)(uintptr_t)(in  + (size_t)blockIdx.z * mat);
  const uint64_t dst = (uint64_t)(uintptr_t)(out + (size_t)blockIdx.z * mat);
  const int r0 = blockIdx.y * 64;
  const int c0 = blockIdx.x * 64;

  // Phase 1: async load  global[r0+r][c0+lx] -> LDS tile[r][lx]   (coalesced)
#pragma unroll
  for (int i = 0; i < 16; ++i) {
    const int r = ly + i * 4;
    const unsigned goff = (unsigned)(((unsigned)(r0 + r) * (unsigned)Ccols +
                                      (unsigned)(c0 + lx)) * 4u);
    const unsigned loff = (unsigned)(uintptr_t)(const void*)&tile[r][lx];
    asm volatile("global_load_async_to_lds_b32 %0, %1, %2"
                 :: "v"(loff), "v"(goff), "s"(src) : "memory");
  }
  asm volatile("s_wait_asynccnt 0x0" ::: "memory");
  __syncthreads();

  // Phase 2: async store LDS tile[lx][c] -> global[c0+c][r0+lx]   (coalesced)
#pragma unroll
  for (int i = 0; i < 16; ++i) {
    const int c = ly + i * 4;
    const unsigned goff = (unsigned)(((unsigned)(c0 + c) * (unsigned)R +
                                      (unsigned)(r0 + lx)) * 4u);
    const unsigned loff = (unsigned)(uintptr_t)(const void*)&tile[lx][c];
    asm volatile("global_store_async_from_lds_b32 %0, %1, %2"
                 :: "v"(goff), "v"(loff), "s"(dst) : "memory");
  }
  asm volatile("s_wait_asynccnt 0x0" ::: "memory");
}

// ---------------------------------------------------------------------------
// Gather kernel (fast path): xt is channel-last (B, HW, C). One block handles
// 4 sites; per site, 64 lanes cover all 256 (b,c) via float4 -> every gather
// is a 16B load from 256 contiguous bytes (L2-resident). Indices/weights are
// block-uniform -> staged once in LDS and broadcast. Output written site-major
// (fully coalesced); final transpose restores (B,C,OH,OW).
// ---------------------------------------------------------------------------
__global__ __launch_bounds__(256) void mmp_gather(const float* __restrict__ xt,
                                                  const int* __restrict__ smap,
                                                  const float* __restrict__ wts,
                                                  float* __restrict__ out_t) {
  __shared__ int   s_idx[4][16];
  __shared__ float s_w[4][16];
  const int t     = threadIdx.x;
  const int site0 = blockIdx.x * 4;

  if (t < 64) {                       // stage 4 sites x 16 (k,p) entries
    const int si = t >> 4;
    const int e  = t & 15;
    const size_t base = (size_t)(site0 + si) * 16 + e;
    int cx = smap[base * 2 + 0];
    int cy = smap[base * 2 + 1];
    cx = cx < 0 ? 0 : (cx > W_ - 1 ? W_ - 1 : cx);
    cy = cy < 0 ? 0 : (cy > H_ - 1 ? H_ - 1 : cy);
    s_idx[si][e] = cy * W_ + cx;
    s_w[si][e]   = wts[base];
  }
  __syncthreads();

  const int si = t >> 6;              // sub-site 0..3
  const int l  = t & 63;              // covers bc: b = l>>4, c-quad = (l&15)*4
  const int b  = l >> 4;
  const int c4 = (l & 15) * 4;
  const float* xb = xt + (size_t)b * (HW_ * C_) + c4;

  const float ninf = -__builtin_inff();
  float4 best = make_float4(ninf, ninf, ninf, ninf);
#pragma unroll
  for (int k = 0; k < 4; ++k) {
    float4 acc = make_float4(0.f, 0.f, 0.f, 0.f);
#pragma unroll
    for (int p = 0; p < 4; ++p) {
      const int e = k * 4 + p;
      const float  wv = s_w[si][e];
      const float4 v  = *(const float4*)(xb + (size_t)s_idx[si][e] * C_);
      acc.x = fmaf(v.x, wv, acc.x);
      acc.y = fmaf(v.y, wv, acc.y);
      acc.z = fmaf(v.z, wv, acc.z);
      acc.w = fmaf(v.w, wv, acc.w);
    }
    best.x = fmaxf(best.x, acc.x);
    best.y = fmaxf(best.y, acc.y);
    best.z = fmaxf(best.z, acc.z);
    best.w = fmaxf(best.w, acc.w);
  }
  *(float4*)(out_t + (size_t)(site0 + si) * BC_ + (size_t)l * 4) = best;
}

// ---------------------------------------------------------------------------
// Fallback (workspace too small): direct gather on original layouts.
// Correct but uncoalesced; only used if ws_size < 168 MB.
// ---------------------------------------------------------------------------
__global__ __launch_bounds__(256) void mmp_gather_direct(const float* __restrict__ x,
                                                         const int* __restrict__ smap,
                                                         const float* __restrict__ wts,
                                                         float* __restrict__ out) {
  __shared__ int   s_idx[16];
  __shared__ float s_w[16];
  const int site = blockIdx.x;
  const int t    = threadIdx.x;       // t == bc index (b*64 + c)

  if (t < 16) {
    const size_t base = (size_t)site * 16 + t;
    int cx = smap[base * 2 + 0];
    int cy = smap[base * 2 + 1];
    cx = cx < 0 ? 0 : (cx > W_ - 1 ? W_ - 1 : cx);
    cy = cy < 0 ? 0 : (cy > H_ - 1 ? H_ - 1 : cy);
    s_idx[t] = cy * W_ + cx;
    s_w[t]   = wts[base];
  }
  __syncthreads();

  const float* xb = x + (size_t)t * HW_;
  float best = -__builtin_inff();
#pragma unroll
  for (int k = 0; k < 4; ++k) {
    float acc = 0.f;
#pragma unroll
    for (int p = 0; p < 4; ++p) {
      const int e = k * 4 + p;
      acc = fmaf(xb[s_idx[e]], s_w[e], acc);
    }
    best = fmaxf(best, acc);
  }
  out[(size_t)t * SITES + site] = best;
}

// ---------------------------------------------------------------------------
extern "C" void kernel_launch(void* const* d_in, const int* in_sizes, int n_in,
                              void* d_out, int out_size, void* d_ws, size_t ws_size,
                              hipStream_t stream) {
  const float* x    = (const float*)d_in[0];
  const int*   smap = (const int*)d_in[1];     // sample_map (int)
  const float* wts  = (const float*)d_in[2];   // interp_weights
  float*       out  = (float*)d_out;
  (void)in_sizes; (void)n_in; (void)out_size;

  const size_t xt_bytes = (size_t)B_ * HW_ * C_ * sizeof(float);   // 134 MB
  const size_t ot_bytes = (size_t)SITES * BC_ * sizeof(float);     // 33.5 MB

  if (d_ws != nullptr && ws_size >= xt_bytes + ot_bytes) {
    float* xt    = (float*)d_ws;
    float* out_t = (float*)((char*)d_ws + xt_bytes);

    // 1) x (B, C, HW) -> xt (B, HW, C): channel-last so gathers are contiguous
    dim3 gA((unsigned)(HW_ / 64), C_ / 64, B_);          // (2048, 1, 4)
    transpose64<<<gA, 256, 0, stream>>>(x, xt, C_, (int)HW_);

    // 2) gather + weighted sum + max, site-major output (coalesced)
    mmp_gather<<<SITES / 4, 256, 0, stream>>>(xt, smap, wts, out_t);

    // 3) out_t (SITES, BC) -> out (BC, SITES) == (B, C, OH, OW)
    dim3 gC(BC_ / 64, SITES / 64, 1);                    // (4, 512, 1)
    transpose64<<<gC, 256, 0, stream>>>(out_t, out, SITES, BC_);
  } else {
    mmp_gather_direct<<<SITES, 256, 0, stream>>>(x, smap, wts, out);
  }
}